// MultiHeadedSelfAttention_43636867727562
// MI455X (gfx1250) — compile-verified
//
#include <hip/hip_runtime.h>
#include <hip/hip_bf16.h>

// Problem constants (from reference): B=16, S=200, D=256, H=8, W=32
#define B_ 16
#define S_ 200
#define D_ 256
#define H_ 8
#define W_ 32

typedef __attribute__((ext_vector_type(2))) float v2f;
typedef __attribute__((ext_vector_type(4))) float f4;
typedef __attribute__((ext_vector_type(8))) float v8f;

// ---------------------------------------------------------------------------
// Kernel 1: QKV projection  Y = X @ W^T + b  via V_WMMA_F32_16X16X4_F32
//   X: (B*S=3200, 256), W: (256,256), Y: (3200,256); 3 matrices (q,k,v).
//   One 16x16 tile per wave; 8 waves per 256-thread block.
//   A 16x4 frag : lanes0-15 row M=l, VGPR0=K(+0|+2), VGPR1=K(+1|+3)
//   B 4x16 frag : lanes0-15 col N=l, VGPR0=K(+0|+2), VGPR1=K(+1|+3)
//   C/D 16x16   : VGPR r -> M = r + (lane<16 ? 0 : 8), N = lane&15
// ---------------------------------------------------------------------------
__global__ __launch_bounds__(256) void qkv_proj_wmma(
    const float* __restrict__ x,
    const float* __restrict__ Wq, const float* __restrict__ bq,
    const float* __restrict__ Wk, const float* __restrict__ bk,
    const float* __restrict__ Wv, const float* __restrict__ bv,
    float* __restrict__ Q, float* __restrict__ K, float* __restrict__ V)
{
    const int MT = (B_ * S_) / 16;   // 200 row tiles
    const int NT = D_ / 16;          // 16 col tiles

    int wave = threadIdx.x >> 5;
    int lane = threadIdx.x & 31;
    int tile = blockIdx.x * 8 + wave;        // 0..9599, grid sized exactly

    int mat = tile / (MT * NT);              // 0=q, 1=k, 2=v (wave-uniform)
    int rem = tile % (MT * NT);
    int tm  = rem / NT;
    int tn  = rem % NT;

    const float* Wm; const float* bias; float* Out;
    if (mat == 0)      { Wm = Wq; bias = bq; Out = Q; }
    else if (mat == 1) { Wm = Wk; bias = bk; Out = K; }
    else               { Wm = Wv; bias = bv; Out = V; }

    int m0   = tm * 16;
    int n0   = tn * 16;
    int half = lane >> 4;        // 0: K+{0,1}, 1: K+{2,3}
    int l15  = lane & 15;
    int ka   = half * 2;

    const float* arow = x  + (size_t)(m0 + l15) * D_ + ka;  // X[m0+l][k0+ka..]
    const float* brow = Wm + (size_t)(n0 + l15) * D_ + ka;  // W^T[k][n] = W[n][k]

    v8f acc = {0.f, 0.f, 0.f, 0.f, 0.f, 0.f, 0.f, 0.f};

    #pragma unroll 4
    for (int k0 = 0; k0 < D_; k0 += 4) {
        v2f a, bf;
        a.x  = arow[k0];     a.y  = arow[k0 + 1];   // contiguous -> b64 load
        bf.x = brow[k0];     bf.y = brow[k0 + 1];
        acc = __builtin_amdgcn_wmma_f32_16x16x4_f32(
                  /*neg_a=*/false, a, /*neg_b=*/false, bf,
                  /*c_mod=*/(short)0, acc, /*reuse_a=*/false, /*reuse_b=*/false);
    }

    float bn    = bias[n0 + l15];
    int   mbase = m0 + half * 8;
    #pragma unroll
    for (int r = 0; r < 8; ++r)
        Out[(size_t)(mbase + r) * D_ + n0 + l15] = acc[r] + bn;
}

// ---------------------------------------------------------------------------
// Kernel 2: fused attention. One workgroup per (b,i); wave h owns head h.
//   scores[h][j] = (q_h . (k_j + tK[i,j]))/sqrt(32) + 10000*(1-mask[j])
//   softmax over j (per wave, via LDS), then
//   out[h][w] = (1/sum) * sum_j e[j] * (v[j][w] + tV[i,j][w])
//   t_K / t_V streamed once with non-temporal b128 loads (512B/wave/iter).
// ---------------------------------------------------------------------------
__global__ __launch_bounds__(256) void fused_attn(
    const float* __restrict__ Q, const float* __restrict__ K,
    const float* __restrict__ V,
    const float* __restrict__ tK, const float* __restrict__ tV,
    const int* __restrict__ mask, float* __restrict__ out)
{
    __shared__ float p_lds[H_][S_];   // 6400 B: unnormalized probabilities
    __shared__ float maskf[S_];       //  800 B: 10000*(1-mask[b][j])

    int bi = blockIdx.x;
    int b  = bi / S_;
    int i  = bi % S_;
    int tid  = threadIdx.x;
    int h    = tid >> 5;              // wave -> head
    int lane = tid & 31;

    if (tid < S_) maskf[tid] = 10000.0f * (1.0f - (float)mask[b * S_ + tid]);
    __syncthreads();

    int jg = lane >> 3;               // 0..3 : j offset within quad
    int dc = (lane & 7) << 2;         // 0,4,..,28 : float offset in head slice
    int dh = h * W_ + dc;             // column within the 256-wide row

    const size_t qoff  = ((size_t)b * S_ + i) * D_;
    const float* tKrow = tK + ((size_t)b * S_ + i) * (size_t)S_ * D_;
    const float* tVrow = tV + ((size_t)b * S_ + i) * (size_t)S_ * D_;
    const float* Kb    = K + (size_t)b * S_ * D_;
    const float* Vb    = V + (size_t)b * S_ * D_;

    f4 q4 = *(const f4*)(Q + qoff + dh);
    const float scale = 0.17677669529663687f;   // 1/sqrt(32)

    // ---- scores + mask -> LDS -------------------------------------------
    for (int j0 = 0; j0 < S_; j0 += 4) {
        int j = j0 + jg;
        f4 t  = __builtin_nontemporal_load((const f4*)(tKrow + (size_t)j * D_ + dh));
        f4 kk = *(const f4*)(Kb + (size_t)j * D_ + dh);
        float s = q4.x * (t.x + kk.x) + q4.y * (t.y + kk.y)
                + q4.z * (t.z + kk.z) + q4.w * (t.w + kk.w);
        // reduce over the 8 lanes sharing this j (bits 0..2)
        s += __shfl_xor(s, 4, 32);
        s += __shfl_xor(s, 2, 32);
        s += __shfl_xor(s, 1, 32);
        if ((lane & 7) == 0)
            p_lds[h][j] = s * scale + maskf[j];
    }

    // ---- per-wave softmax over 200 entries ------------------------------
    float mx = -3.4e38f;
    for (int j = lane; j < S_; j += 32) mx = fmaxf(mx, p_lds[h][j]);
    #pragma unroll
    for (int off = 16; off; off >>= 1) mx = fmaxf(mx, __shfl_xor(mx, off, 32));

    float sum = 0.f;
    for (int j = lane; j < S_; j += 32) {
        float e = __expf(p_lds[h][j] - mx);
        p_lds[h][j] = e;
        sum += e;
    }
    #pragma unroll
    for (int off = 16; off; off >>= 1) sum += __shfl_xor(sum, off, 32);
    float inv = 1.0f / sum;

    // ---- output: p @ (v + tV), streaming t_V ----------------------------
    f4 acc = {0.f, 0.f, 0.f, 0.f};
    for (int j0 = 0; j0 < S_; j0 += 4) {
        int j = j0 + jg;
        float pj = p_lds[h][j];
        f4 tv = __builtin_nontemporal_load((const f4*)(tVrow + (size_t)j * D_ + dh));
        f4 vv = *(const f4*)(Vb + (size_t)j * D_ + dh);
        acc.x = fmaf(pj, tv.x + vv.x, acc.x);
        acc.y = fmaf(pj, tv.y + vv.y, acc.y);
        acc.z = fmaf(pj, tv.z + vv.z, acc.z);
        acc.w = fmaf(pj, tv.w + vv.w, acc.w);
    }
    // reduce across the 4 j-groups (bits 3,4)
    acc.x += __shfl_xor(acc.x, 8, 32);  acc.x += __shfl_xor(acc.x, 16, 32);
    acc.y += __shfl_xor(acc.y, 8, 32);  acc.y += __shfl_xor(acc.y, 16, 32);
    acc.z += __shfl_xor(acc.z, 8, 32);  acc.z += __shfl_xor(acc.z, 16, 32);
    acc.w += __shfl_xor(acc.w, 8, 32);  acc.w += __shfl_xor(acc.w, 16, 32);

    if (lane < 8) {
        f4 o = acc * inv;
        *(f4*)(out + qoff + dh) = o;    // out[b,i,h*32+dc .. +3]
    }
}

// ---------------------------------------------------------------------------
// Launcher. Inputs (setup_inputs order):
//   0:x 1:t_K 2:t_V 3:mask 4:Wq 5:bq 6:Wk 7:bk 8:Wv 9:bv 10:n_heads
// Workspace: Q,K,V fp32 (3 * 3200*256 * 4B = 9.83 MB)
// ---------------------------------------------------------------------------
extern "C" void kernel_launch(void* const* d_in, const int* in_sizes, int n_in,
                              void* d_out, int out_size, void* d_ws, size_t ws_size,
                              hipStream_t stream) {
    const float* x    = (const float*)d_in[0];
    const float* t_K  = (const float*)d_in[1];
    const float* t_V  = (const float*)d_in[2];
    const int*   mask = (const int*)d_in[3];
    const float* Wq   = (const float*)d_in[4];
    const float* bq   = (const float*)d_in[5];
    const float* Wk   = (const float*)d_in[6];
    const float* bk   = (const float*)d_in[7];
    const float* Wv   = (const float*)d_in[8];
    const float* bv   = (const float*)d_in[9];
    float*       outp = (float*)d_out;

    size_t n = (size_t)B_ * S_ * D_;
    float* Q = (float*)d_ws;
    float* K = Q + n;
    float* V = K + n;

    // 3 matrices * 200 * 16 tiles = 9600 tiles, 8 waves/block -> 1200 blocks
    qkv_proj_wmma<<<1200, 256, 0, stream>>>(x, Wq, bq, Wk, bk, Wv, bv, Q, K, V);
    // one workgroup per (b,i)
    fused_attn<<<B_ * S_, 256, 0, stream>>>(Q, K, V, t_K, t_V, mask, outp);
}